// Temporal_63187558858771
// MI455X (gfx1250) — compile-verified
//
#include <hip/hip_runtime.h>

typedef __attribute__((ext_vector_type(2))) float v2f;
typedef __attribute__((ext_vector_type(8))) float v8f;

#define NN 1024   // inputs per row
#define OO 1024   // outputs
#define BB 64     // batch
#define NO_SPIKE 1e9f

// ---------------------------------------------------------------------------
// Kernel 1: transpose weight [O][N] -> Wt [N][O] so that the sorted-order
// gather becomes coalesced (Wt[idx][j..j+15] is contiguous).
// ---------------------------------------------------------------------------
__global__ void transpose_w(const float* __restrict__ w, float* __restrict__ wt) {
    __shared__ float tile[32][33];
    const int x0 = blockIdx.x * 32;   // N dim
    const int y0 = blockIdx.y * 32;   // O dim
    const int tx = threadIdx.x;       // 0..31
    const int ty = threadIdx.y;       // 0..7
#pragma unroll
    for (int r = 0; r < 32; r += 8)
        tile[ty + r][tx] = w[(size_t)(y0 + ty + r) * NN + (x0 + tx)];
    __syncthreads();
#pragma unroll
    for (int r = 0; r < 32; r += 8)
        wt[(size_t)(x0 + ty + r) * OO + (y0 + tx)] = tile[tx][ty + r];
}

// ---------------------------------------------------------------------------
// Kernel 2: per-batch bitonic sort of (z, original index), ascending z.
// One block of 512 threads per batch row; 1024 elements in LDS.
// ---------------------------------------------------------------------------
__global__ void sort_rows(const float* __restrict__ x,
                          float* __restrict__ zs, int* __restrict__ od) {
    __shared__ float sz[NN];
    __shared__ int   si[NN];
    const int b = blockIdx.x;
    for (int i = threadIdx.x; i < NN; i += blockDim.x) {
        sz[i] = x[(size_t)b * NN + i];
        si[i] = i;
    }
    __syncthreads();
    for (int k = 2; k <= NN; k <<= 1) {
        for (int j = k >> 1; j > 0; j >>= 1) {
            for (int i = threadIdx.x; i < NN; i += blockDim.x) {
                const int ixj = i ^ j;
                if (ixj > i) {
                    const bool up = ((i & k) == 0);
                    const float a = sz[i], c = sz[ixj];
                    if (up ? (a > c) : (a < c)) {
                        sz[i] = c; sz[ixj] = a;
                        const int t = si[i]; si[i] = si[ixj]; si[ixj] = t;
                    }
                }
            }
            __syncthreads();
        }
    }
    for (int i = threadIdx.x; i < NN; i += blockDim.x) {
        zs[(size_t)b * NN + i] = sz[i];
        od[(size_t)b * NN + i] = si[i];
    }
}

// ---------------------------------------------------------------------------
// Kernel 3: WMMA prefix-sum scan + first-valid search.
// One wave (32 lanes) per (batch, 16-output tile). For each 16-wide k-tile:
//   cw_tile  = sum_s A_s(w)   x T_s   (4 chained V_WMMA_F32_16X16X4_F32)
//   cwz_tile = sum_s A_s(w*z) x T_s   (4 more)
// where T_s[r,k] = (4s+r <= k), giving an exact f32 in-tile cumulative sum.
// Validity test is division-free: with denom>0,
//   zk <= cwz/denom < znext  <=>  cwz >= zk*denom && cwz < znext*denom.
// The single division per found output happens once, after the scan.
// ---------------------------------------------------------------------------
__global__ void ttfs_scan(const float* __restrict__ wt,
                          const float* __restrict__ zs,
                          const int*   __restrict__ od,
                          const int*   __restrict__ Qp,
                          float* __restrict__ out) {
    __shared__ float lz[NN];
    __shared__ int   lo[NN];
    const int b   = blockIdx.x;
    const int tid = threadIdx.x;            // 0..255 (8 waves)
    for (int i = tid; i < NN; i += 256) {
        lz[i] = zs[(size_t)b * NN + i];
        lo[i] = od[(size_t)b * NN + i];
    }
    __syncthreads();

    const float Qf   = (float)Qp[0];
    const int wave   = tid >> 5;
    const int lane   = tid & 31;
    const int m      = lane & 15;           // A row / B,D column within tile
    const int hi     = lane >> 4;           // half-wave select
    const int jt     = blockIdx.y * 8 + wave;
    const int jbase  = jt * 16;
    const float* wcol = wt + (jbase + m);   // this lane's output-neuron column in Wt

    float base_cw[8], base_cz[8], bn[8], bd[8];
    int   bk[8];
#pragma unroll
    for (int c = 0; c < 8; ++c) {
        base_cw[c] = 0.f; base_cz[c] = 0.f;
        bk[c] = 0x7FFFFFFF; bn[c] = NO_SPIKE; bd[c] = 1.f;
    }

    const int srcl = (lane & 16) | 15;      // column-15 lane of my half (15 or 31)

    for (int t = 0; t < NN / 16; ++t) {
        const int tbase = t * 16;
        v8f cw = {}, cz = {};
#pragma unroll
        for (int s = 0; s < 4; ++s) {
            const int r0 = 4 * s + 2 * hi;  // within-tile K position of a.x
            const int r1 = r0 + 1;          // within-tile K position of a.y
            const int p0 = tbase + r0, p1 = tbase + r1;
            const int i0 = lo[p0],     i1 = lo[p1];
            const float z0 = lz[p0],   z1 = lz[p1];
            v2f a, az, bm;
            a.x  = wcol[(size_t)i0 * OO];   // w[jbase+m, sorted pos p0]
            a.y  = wcol[(size_t)i1 * OO];
            az.x = a.x * z0;
            az.y = a.y * z1;
            bm.x = (r0 <= m) ? 1.f : 0.f;   // step-triangular mask T_s
            bm.y = (r1 <= m) ? 1.f : 0.f;
            cw = __builtin_amdgcn_wmma_f32_16x16x4_f32(false, a,  false, bm, (short)0, cw, false, false);
            cz = __builtin_amdgcn_wmma_f32_16x16x4_f32(false, az, false, bm, (short)0, cz, false, false);
        }

        // Validity test per element of the 16x16 tile (column = m, rows = c+8*hi).
        const int   pos   = tbase + m;
        const float zk    = lz[pos];
        const float znext = (pos < NN - 1) ? lz[pos + 1] : NO_SPIKE;
#pragma unroll
        for (int c = 0; c < 8; ++c) {
            const float cwf   = base_cw[c] + cw[c];
            const float czf   = base_cz[c] + cz[c];
            const float denom = cwf - Qf;
            const bool valid  = (denom > 0.f) &&
                                (czf >= zk * denom) && (czf < znext * denom);
            if (valid && pos < bk[c]) { bk[c] = pos; bn[c] = czf; bd[c] = denom; }
            // carry running bases forward using the column-15 tile totals
            base_cw[c] += __shfl(cw[c], srcl, 32);
            base_cz[c] += __shfl(cz[c], srcl, 32);
        }

        // Wave-uniform early exit: every output row has found a candidate.
        unsigned fmask = 0;
#pragma unroll
        for (int c = 0; c < 8; ++c) fmask |= (bk[c] < NN) ? (1u << c) : 0u;
#pragma unroll
        for (int d = 1; d < 16; d <<= 1) fmask |= (unsigned)__shfl_xor((int)fmask, d, 32);
        const unsigned other = (unsigned)__shfl_xor((int)fmask, 16, 32);
        if ((fmask & other) == 0xFFu) break;
    }

    // One division per tracked candidate, then min-position reduction across
    // the 16 column-lanes of each half.
#pragma unroll
    for (int c = 0; c < 8; ++c) {
        int   k = bk[c];
        float z = (k < NN) ? (bn[c] / bd[c]) : NO_SPIKE;
#pragma unroll
        for (int d = 1; d < 16; d <<= 1) {
            const int   ko = __shfl_xor(k, d, 32);
            const float zo = __shfl_xor(z, d, 32);
            if (ko < k) { k = ko; z = zo; }
        }
        bk[c] = k; bn[c] = z;
    }
    if (m == 0) { // lanes 0 and 16 write rows 0-7 and 8-15 respectively
        const int jrow = jbase + 8 * hi;
#pragma unroll
        for (int c = 0; c < 8; ++c)
            out[(size_t)b * OO + jrow + c] = (bk[c] < NN) ? bn[c] : NO_SPIKE;
    }
}

// ---------------------------------------------------------------------------
extern "C" void kernel_launch(void* const* d_in, const int* in_sizes, int n_in,
                              void* d_out, int out_size, void* d_ws, size_t ws_size,
                              hipStream_t stream) {
    const float* x = (const float*)d_in[0];          // [B,N]
    const float* w = (const float*)d_in[1];          // [O,N]
    const int*   Q = (const int*)d_in[2];            // scalar
    float* out = (float*)d_out;                      // [B,O]

    float* wt = (float*)d_ws;                        // [N,O]  4 MB
    float* zsort = wt + (size_t)NN * OO;             // [B,N]  256 KB
    int*   order = (int*)(zsort + (size_t)BB * NN);  // [B,N]  256 KB

    hipLaunchKernelGGL(transpose_w, dim3(NN / 32, OO / 32), dim3(32, 8), 0, stream, w, wt);
    hipLaunchKernelGGL(sort_rows, dim3(BB), dim3(512), 0, stream, x, zsort, order);
    hipLaunchKernelGGL(ttfs_scan, dim3(BB, OO / 128), dim3(256), 0, stream,
                       wt, zsort, order, Q, out);
}